// ReduceMaxGrad_53833120088407
// MI455X (gfx1250) — compile-verified
//
#include <hip/hip_runtime.h>
#include <cstdint>

// Problem geometry (from reference):
//   grad_last : (B=8, 4096) f32
//   indices   : (4096,) int32 (JAX default x64-disabled downcasts int64->int32;
//               harness instructs integer -> const int*)
//   out       : (8, 4096, 2048) f32, out[b,i,j] = (j==idx[i]) ? grad[b,i] : 0
//
// Pure store-bandwidth problem: 256 MB of output, ~0 FLOPs.
// Strategy: per-wave async zero-fill (GLOBAL_STORE_ASYNC_FROM_LDS_B128 from a
// 512 B LDS zero buffer, non-temporal since 256 MB > 192 MB L2), then
// s_wait_asynccnt 0 and a single regular store to patch the one-hot element.

#define ROW_LEN   2048                   // floats per output row
#define ROW_BYTES (ROW_LEN * 4)         // 8192 B per row
#define ROWS      (8 * 4096)            // 32768 rows total
#define WAVES_PER_BLOCK 8               // 256 threads = 8 wave32

__global__ __launch_bounds__(256)
void OneHotScatterGrad_kernel(const float* __restrict__ grad,
                              const int*   __restrict__ idx,
                              float*       __restrict__ out) {
    // 512 B of zeros in LDS, shared (read-only) by all 8 waves' async stores.
    __shared__ float4 zbuf[32];
    const int tid = threadIdx.x;
    if (tid < 32) zbuf[tid] = float4{0.f, 0.f, 0.f, 0.f};
    __syncthreads();   // DS writes complete + barrier before async engine reads LDS

    const int lane = tid & 31;
    const int wave = tid >> 5;
    const int row  = blockIdx.x * WAVES_PER_BLOCK + wave;   // 0 .. 32767
    if (row >= ROWS) return;

    // Workgroup-relative LDS byte offset of this lane's 16 B of zeros.
    // Generic LDS pointer low 32 bits == LDS offset (flat-LDS aperture mapping).
    const uint32_t lds_off = (uint32_t)(uintptr_t)(&zbuf[lane]);

    // Zero-fill this row: 2048 f32 = 8 KB = 16 iterations x (32 lanes x 16 B).
    // saddr form: 64-bit base in SGPR pair, 32-bit voffset in VGPR
    // (output is 2^28 B total, fits signed-32 offset).
    const uint32_t row_base = (uint32_t)row * (uint32_t)ROW_BYTES;
#pragma unroll
    for (int it = 0; it < 16; ++it) {
        const uint32_t voff = row_base + (uint32_t)(it * 512 + lane * 16);
        asm volatile(
            "global_store_async_from_lds_b128 %0, %1, %2 th:TH_STORE_NT"
            :
            : "v"(voff), "v"(lds_off), "s"(out)
            : "memory");
    }

    // Wave-local async stores must land before we overwrite one element of the
    // same row with a regular store (different counters => otherwise unordered).
    __builtin_amdgcn_s_wait_asynccnt(0);

    if (lane == 0) {
        const int i = row & 4095;                 // row = b*4096 + i
        const int j = idx[i];                     // 0 .. 2047
        out[(size_t)row * ROW_LEN + (size_t)j] = grad[row];
    }
}

extern "C" void kernel_launch(void* const* d_in, const int* in_sizes, int n_in,
                              void* d_out, int out_size, void* d_ws, size_t ws_size,
                              hipStream_t stream) {
    (void)in_sizes; (void)n_in; (void)d_ws; (void)ws_size; (void)out_size;
    const float* grad = (const float*)d_in[0];   // (8, 4096) f32
    const int*   idx  = (const int*)d_in[1];     // (4096,) int32
    float*       out  = (float*)d_out;           // (8, 4096, 2048) f32

    const int blocks = ROWS / WAVES_PER_BLOCK;   // 4096
    OneHotScatterGrad_kernel<<<blocks, 256, 0, stream>>>(grad, idx, out);
}